// Net_3564822856025
// MI455X (gfx1250) — compile-verified
//
#include <hip/hip_runtime.h>
#include <math.h>

#define NNODES 100000
#define FIN    1433
#define HID1   32
#define HID2   16
#define NEDGES 3200000

typedef __attribute__((ext_vector_type(2))) float v2f;
typedef __attribute__((ext_vector_type(8))) float v8f;

__device__ __forceinline__ void atomAddF(float* p, float v) {
    // Relaxed agent-scope fp32 atomic add, no return -> global_atomic_add_f32
    __hip_atomic_fetch_add(p, v, __ATOMIC_RELAXED, __HIP_MEMORY_SCOPE_AGENT);
}

// ---------------- utility kernels ----------------
__global__ void k_fill(float* __restrict__ p, float val, long n) {
    long i = (long)blockIdx.x * blockDim.x + threadIdx.x;
    if (i < n) p[i] = val;
}

__global__ void k_deg_count(const int* __restrict__ dst, float* __restrict__ deg) {
    long e = (long)blockIdx.x * blockDim.x + threadIdx.x;
    if (e < NEDGES) atomAddF(&deg[dst[e]], 1.0f);
}

__global__ void k_rsqrt_inplace(float* __restrict__ d, int n) {
    int i = blockIdx.x * blockDim.x + threadIdx.x;
    if (i < n) {
        float v = d[i];
        d[i] = (v > 0.0f) ? rsqrtf(v) : 0.0f;
    }
}

// ---------------- fp32 WMMA GEMM: C[M,NC] = A[M,K] @ B[K,NC] ----------------
// One wave computes a 16-row strip and all NC (16 or 32) columns using
// V_WMMA_F32_16X16X4_F32. M must be a multiple of 16 (100000 = 6250*16).
// Main loop over K&~3 is branch-free (no EXEC masking); the K%4 remainder is
// folded in with scalar FMAs directly into the striped accumulator.
template <int NC>
__global__ void k_gemm_wmma(const float* __restrict__ A, const float* __restrict__ B,
                            float* __restrict__ C, int M, int K) {
    const int  lane  = threadIdx.x & 31;
    const int  wave  = threadIdx.x >> 5;
    const long strip = (long)blockIdx.x * (blockDim.x >> 5) + wave;
    const long row0  = strip * 16;
    if (row0 >= M) return;            // uniform per wave -> EXEC all-ones inside

    const int  m    = lane & 15;      // M index for A frag / N index for B frag
    const int  kh   = lane >> 4;      // which K-pair this half-wave holds
    const float* __restrict__ Arow = A + (row0 + m) * (long)K;

    v8f acc0 = {};
    v8f acc1 = {};

    const int Kmain = K & ~3;
#pragma unroll 2
    for (int k = 0; k < Kmain; k += 4) {
        const int ka = k + kh * 2;
        v2f a = { Arow[ka], Arow[ka + 1] };
        v2f b = { B[(long)ka * NC + m], B[(long)(ka + 1) * NC + m] };
        acc0 = __builtin_amdgcn_wmma_f32_16x16x4_f32(
                   false, a, false, b, (short)0, acc0, false, false);
        if (NC == 32) {
            v2f bb = { B[(long)ka * NC + 16 + m], B[(long)(ka + 1) * NC + 16 + m] };
            acc1 = __builtin_amdgcn_wmma_f32_16x16x4_f32(
                       false, a, false, bb, (short)0, acc1, false, false);
        }
    }

    // K % 4 tail: D[row0 + r + 8*kh][m] += A[...][k] * B[k][m]
    for (int k = Kmain; k < K; ++k) {
        float bv0 = B[(long)k * NC + m];
        float bv1 = (NC == 32) ? B[(long)k * NC + 16 + m] : 0.0f;
#pragma unroll
        for (int r = 0; r < 8; ++r) {
            float av = A[(row0 + r + kh * 8) * (long)K + k];
            acc0[r] = fmaf(av, bv0, acc0[r]);
            if (NC == 32) acc1[r] = fmaf(av, bv1, acc1[r]);
        }
    }

    // D layout: VGPR r -> row = row0 + r + 8*kh, col = lane&15
#pragma unroll
    for (int r = 0; r < 8; ++r) {
        long orow = row0 + r + kh * 8;
        C[orow * NC + m] = acc0[r];
        if (NC == 32) C[orow * NC + 16 + m] = acc1[r];
    }
}

// ---------------- edge scatter: out[dst] += norm * h[src] ----------------
// One thread per (edge, feature): consecutive lanes cover the H features of
// one edge -> coalesced gather + contiguous fp32 atomics (L2-resident).
template <int H>
__global__ void k_scatter(const int* __restrict__ src, const int* __restrict__ dst,
                          const float* __restrict__ dis, const float* __restrict__ hin,
                          float* __restrict__ hout) {
    long t = (long)blockIdx.x * blockDim.x + threadIdx.x;
    long e = t / H;                   // H is a power of two -> shift
    int  f = (int)(t - e * H);
    if (e >= NEDGES) return;
    int s = src[e];
    int d = dst[e];
    float norm = dis[s] * dis[d];
    atomAddF(&hout[(long)d * H + f], norm * hin[(long)s * H + f]);
}

// ---------------- fused self-loop + bias + ReLU ----------------
// out[i,f] = relu(out[i,f] + deg_isqrt[i]^2 * h[i,f] + b[f])
template <int H>
__global__ void k_self_bias_relu(const float* __restrict__ dis, const float* __restrict__ ht,
                                 const float* __restrict__ b, float* __restrict__ h) {
    long t = (long)blockIdx.x * blockDim.x + threadIdx.x;
    if (t >= (long)NNODES * H) return;
    long i = t / H;
    int  f = (int)(t - i * H);
    float di = dis[i];
    float v  = h[t] + di * di * ht[t] + b[f];
    h[t] = (v > 0.0f) ? v : 0.0f;
}

// ---------------- final: out = log_softmax(h2 @ Wl + bl) ----------------
// One 256-thread block per node; logits staged in LDS; wave32 reductions.
// Output is written with non-temporal stores (573 MB streamed once).
__global__ void k_final(const float* __restrict__ h2, const float* __restrict__ Wl,
                        const float* __restrict__ bl, float* __restrict__ out) {
    __shared__ float sh_h[HID2];
    __shared__ float sh_logit[FIN];
    __shared__ float sred[16];

    const int  tid  = threadIdx.x;
    const long node = blockIdx.x;

    if (tid < HID2) sh_h[tid] = h2[node * HID2 + tid];
    __syncthreads();

    float lmax = -3.402823466e38f;
    for (int c = tid; c < FIN; c += blockDim.x) {
        float v = bl[c];
#pragma unroll
        for (int k = 0; k < HID2; ++k) v = fmaf(sh_h[k], Wl[(long)k * FIN + c], v);
        sh_logit[c] = v;
        lmax = fmaxf(lmax, v);
    }
#pragma unroll
    for (int off = 16; off > 0; off >>= 1)
        lmax = fmaxf(lmax, __shfl_down(lmax, off, 32));
    if ((tid & 31) == 0) sred[tid >> 5] = lmax;
    __syncthreads();
    if (tid == 0) {
        float v = sred[0];
        for (int w = 1; w < (int)(blockDim.x >> 5); ++w) v = fmaxf(v, sred[w]);
        sred[0] = v;
    }
    __syncthreads();
    const float gmax = sred[0];

    float lsum = 0.0f;
    for (int c = tid; c < FIN; c += blockDim.x) lsum += expf(sh_logit[c] - gmax);
#pragma unroll
    for (int off = 16; off > 0; off >>= 1) lsum += __shfl_down(lsum, off, 32);
    if ((tid & 31) == 0) sred[8 + (tid >> 5)] = lsum;
    __syncthreads();
    if (tid == 0) {
        float v = 0.0f;
        for (int w = 0; w < (int)(blockDim.x >> 5); ++w) v += sred[8 + w];
        sred[8] = logf(v);
    }
    __syncthreads();
    const float lz = sred[8];

    for (int c = tid; c < FIN; c += blockDim.x)
        __builtin_nontemporal_store(sh_logit[c] - gmax - lz, &out[node * FIN + c]);
}

// ---------------- launcher ----------------
extern "C" void kernel_launch(void* const* d_in, const int* in_sizes, int n_in,
                              void* d_out, int out_size, void* d_ws, size_t ws_size,
                              hipStream_t stream) {
    (void)in_sizes; (void)n_in; (void)out_size; (void)ws_size;

    const float* x  = (const float*)d_in[0];
    const float* W1 = (const float*)d_in[1];
    const float* b1 = (const float*)d_in[2];
    const float* W2 = (const float*)d_in[3];
    const float* b2 = (const float*)d_in[4];
    const float* Wl = (const float*)d_in[5];
    const float* bl = (const float*)d_in[6];
    const int*   ei = (const int*)d_in[7];   // [2, E] int32 (JAX x64 disabled)
    const int* src = ei;
    const int* dst = ei + NEDGES;

    float* ws  = (float*)d_ws;
    float* dis = ws;                          // N        (deg -> deg^-1/2)
    float* h1t = dis + NNODES;                // N*32     (x @ W1)
    float* h1  = h1t + (long)NNODES * HID1;   // N*32     (post-aggregation)
    float* h2t = h1  + (long)NNODES * HID1;   // N*16
    float* h2  = h2t + (long)NNODES * HID2;   // N*16
    float* out = (float*)d_out;

    const int TB = 256;

    // degrees (self-loop counted as the initial 1.0)
    k_fill<<<(NNODES + TB - 1) / TB, TB, 0, stream>>>(dis, 1.0f, (long)NNODES);
    k_deg_count<<<(NEDGES + TB - 1) / TB, TB, 0, stream>>>(dst, dis);
    k_rsqrt_inplace<<<(NNODES + TB - 1) / TB, TB, 0, stream>>>(dis, NNODES);

    // layer 1: h1t = x @ W1  (WMMA f32)
    {
        long strips = (NNODES + 15) / 16;
        int  grid   = (int)((strips + 7) / 8);   // 8 waves / block
        k_gemm_wmma<HID1><<<grid, TB, 0, stream>>>(x, W1, h1t, NNODES, FIN);
    }
    k_fill<<<(int)(((long)NNODES * HID1 + TB - 1) / TB), TB, 0, stream>>>(h1, 0.0f, (long)NNODES * HID1);
    k_scatter<HID1><<<(int)(((long)NEDGES * HID1 + TB - 1) / TB), TB, 0, stream>>>(src, dst, dis, h1t, h1);
    k_self_bias_relu<HID1><<<(int)(((long)NNODES * HID1 + TB - 1) / TB), TB, 0, stream>>>(dis, h1t, b1, h1);

    // layer 2: h2t = h1 @ W2  (WMMA f32)
    {
        long strips = (NNODES + 15) / 16;
        int  grid   = (int)((strips + 7) / 8);
        k_gemm_wmma<HID2><<<grid, TB, 0, stream>>>(h1, W2, h2t, NNODES, HID1);
    }
    k_fill<<<(int)(((long)NNODES * HID2 + TB - 1) / TB), TB, 0, stream>>>(h2, 0.0f, (long)NNODES * HID2);
    k_scatter<HID2><<<(int)(((long)NEDGES * HID2 + TB - 1) / TB), TB, 0, stream>>>(src, dst, dis, h2t, h2);
    k_self_bias_relu<HID2><<<(int)(((long)NNODES * HID2 + TB - 1) / TB), TB, 0, stream>>>(dis, h2t, b2, h2);

    // final linear + log_softmax, fused per node
    k_final<<<NNODES, 256, 0, stream>>>(h2, Wl, bl, out);
}